// SingleT2FLS_Mamdani_78769700208732
// MI455X (gfx1250) — compile-verified
//
#include <hip/hip_runtime.h>
#include <hip/hip_bf16.h>
#include <math.h>

// SingleT2FLS Mamdani (interval type-2 fuzzy, Karnik-Mendel) on gfx1250.
// Strategy:
//  - One fused kernel. Per-block prep in LDS: stable insertion-argsort of
//    c1/c2 (16 elems), rule-parameter tables in sorted order, and 6
//    triangular weight matrices.
//  - Per wave: 16 samples. The KM cumulative sums collapse to
//    s[r] = sum_k c1s[k] * (k<=r ? UU_sorted[k] : LL_sorted[k]),
//    i.e. triangular matmuls over the 16-rule axis -> chained
//    V_WMMA_F32_16X16X4_F32 (4 K-chunks per product, 32 WMMAs / 16 samples).
//  - cummin/argmin + final weighted averages done with per-lane scans over
//    the C/D layout (rows 0-7 in lanes 0-15, rows 8-15 in lanes 16-31) plus
//    shfl_xor(16) pair combines.
// Operand layouts (per CDNA5 ISA 7.12.2):
//  A f32 16x4 : lane holds A[M=lane&15][K=2*(lane>>4)+v], v=VGPR 0/1
//  B f32 4x16 : lane holds B[K=2*(lane>>4)+v][N=lane&15]
//  C/D f32 16x16: VGPR v holds row M=v (lanes 0-15) / M=8+v (lanes 16-31),
//                 column N = lane&15.

#define RULES 16
#define ANTE  4

typedef float v8f __attribute__((ext_vector_type(8)));
typedef float v2f __attribute__((ext_vector_type(2)));

static __device__ __forceinline__ v8f wmma4(v2f a, v2f b, v8f c) {
  // (neg_a, A, neg_b, B, c_mod, C, reuse_a, reuse_b)
  return __builtin_amdgcn_wmma_f32_16x16x4_f32(false, a, false, b,
                                               (short)0, c, false, false);
}

__global__ __launch_bounds__(256) void t2fls_km_kernel(
    const float* __restrict__ input, const float* __restrict__ frb,
    const float* __restrict__ c1g, const float* __restrict__ c2g,
    float* __restrict__ out, int N)
{
  __shared__ float sP1[RULES * 12];          // sorted-by-c1: m[4], 0.5/shi^2[4], 0.5/slo^2[4]
  __shared__ float sP2[RULES * 12];          // sorted-by-c2
  __shared__ float sW[6][256];               // 0:c1s*incl 1:c1s*excl 2:c2s*incl 3:c2s*excl 4:incl 5:excl
  __shared__ float sC1s[16], sC1[16], sC2s[16], sC2[16];
  __shared__ int   sI1[16], sI2[16];

  const int tid = threadIdx.x;

  if (tid == 0) {                            // stable ascending argsort (matches jnp.argsort)
    int i1[16], i2[16];
    for (int r = 0; r < 16; ++r) { i1[r] = r; i2[r] = r; }
    for (int i = 1; i < 16; ++i) {
      int k = i1[i]; float v = c1g[k]; int j = i - 1;
      for (; j >= 0 && c1g[i1[j]] > v; --j) i1[j + 1] = i1[j];
      i1[j + 1] = k;
    }
    for (int i = 1; i < 16; ++i) {
      int k = i2[i]; float v = c2g[k]; int j = i - 1;
      for (; j >= 0 && c2g[i2[j]] > v; --j) i2[j + 1] = i2[j];
      i2[j + 1] = k;
    }
    for (int r = 0; r < 16; ++r) {
      sI1[r] = i1[r];        sI2[r] = i2[r];
      sC1s[r] = c1g[i1[r]];  sC2s[r] = c2g[i2[r]];
      sC1[r] = c1g[r];       sC2[r] = c2g[r];
    }
  }
  __syncthreads();

  { // 6 triangular weight matrices, one (r,k) entry per thread
    int r = tid >> 4, k = tid & 15, e = r * 16 + k;
    float inc = (k <= r) ? 1.0f : 0.0f;
    float exc = 1.0f - inc;
    sW[0][e] = sC1s[k] * inc;  sW[1][e] = sC1s[k] * exc;
    sW[2][e] = sC2s[k] * inc;  sW[3][e] = sC2s[k] * exc;
    sW[4][e] = inc;            sW[5][e] = exc;
  }
  if (tid < 32) { // rule-parameter tables in sorted order
    int kk = tid & 15;
    int rule = (tid < 16) ? sI1[kk] : sI2[kk];
    float* P = (tid < 16) ? sP1 : sP2;
    #pragma unroll
    for (int a = 0; a < 4; ++a) {
      float m  = frb[rule * 12 + a * 3 + 0];
      float sa = frb[rule * 12 + a * 3 + 1];
      float sb = frb[rule * 12 + a * 3 + 2];
      float slo = fminf(sa, sb), shi = fmaxf(sa, sb);
      P[kk * 12 + a]     = m;
      P[kk * 12 + 4 + a] = 0.5f / (shi * shi);
      P[kk * 12 + 8 + a] = 0.5f / (slo * slo);
    }
  }
  __syncthreads();

  const int lane = tid & 31;
  const int n = lane & 15;                   // sample column of this lane
  const int h = lane >> 4;                   // lane-half (K/M split)
  const int wave = blockIdx.x * (blockDim.x >> 5) + (tid >> 5);
  const int samp = wave * 16 + n;
  const int sc = samp < N ? samp : N - 1;    // clamp keeps EXEC uniform for WMMA

  const float4 xv = *(const float4*)(input + (size_t)sc * 4);

  // B operands: memberships for this lane's 8 rule slots kk = 4c + 2h + v
  float Bu1[8], Bl1[8], Bu2[8], Bl2[8];
  #pragma unroll
  for (int c = 0; c < 4; ++c) {
    #pragma unroll
    for (int v = 0; v < 2; ++v) {
      const int j = c * 2 + v;
      const int kk = 4 * c + 2 * h + v;
      {
        const float* P = sP1 + kk * 12;
        float z0 = xv.x - P[0], z1 = xv.y - P[1], z2 = xv.z - P[2], z3 = xv.w - P[3];
        float qh = z0*z0*P[4] + z1*z1*P[5] + z2*z2*P[6] + z3*z3*P[7];
        float ql = z0*z0*P[8] + z1*z1*P[9] + z2*z2*P[10] + z3*z3*P[11];
        Bu1[j] = __expf(-qh);                // product of 4 Gaussians fused into one exp
        Bl1[j] = __expf(-ql);
      }
      {
        const float* P = sP2 + kk * 12;
        float z0 = xv.x - P[0], z1 = xv.y - P[1], z2 = xv.z - P[2], z3 = xv.w - P[3];
        float qh = z0*z0*P[4] + z1*z1*P[5] + z2*z2*P[6] + z3*z3*P[7];
        float ql = z0*z0*P[8] + z1*z1*P[9] + z2*z2*P[10] + z3*z3*P[11];
        Bu2[j] = __expf(-qh);
        Bl2[j] = __expf(-ql);
      }
    }
  }

  // Triangular matmuls: S/T (left), SR/TR (right), chained over 4 K-chunks.
  v8f S = {}; v8f T = {}; v8f SR = {}; v8f TR = {};
  #pragma unroll
  for (int c = 0; c < 4; ++c) {
    const int b0 = n * 16 + 4 * c + 2 * h;
    v2f aSI = { sW[0][b0], sW[0][b0 + 1] };
    v2f aSE = { sW[1][b0], sW[1][b0 + 1] };
    v2f aRI = { sW[2][b0], sW[2][b0 + 1] };
    v2f aRE = { sW[3][b0], sW[3][b0 + 1] };
    v2f aTI = { sW[4][b0], sW[4][b0 + 1] };
    v2f aTE = { sW[5][b0], sW[5][b0 + 1] };
    v2f bU1 = { Bu1[2*c], Bu1[2*c+1] };
    v2f bL1 = { Bl1[2*c], Bl1[2*c+1] };
    v2f bU2 = { Bu2[2*c], Bu2[2*c+1] };
    v2f bL2 = { Bl2[2*c], Bl2[2*c+1] };
    S  = wmma4(aSI, bU1, S );  S  = wmma4(aSE, bL1, S );   // s[r]
    T  = wmma4(aTI, bU1, T );  T  = wmma4(aTE, bL1, T );   // t[r]
    SR = wmma4(aRI, bL2, SR);  SR = wmma4(aRE, bU2, SR);   // sr[r]
    TR = wmma4(aTI, bL2, TR);  TR = wmma4(aTE, bU2, TR);   // tr[r]
  }

  // s0 = sum c1s*LLs, t0 = sum LLs ; s0r = sum c2s*UUr, t0r = sum UUr
  float s0p = 0, t0p = 0, s0rp = 0, t0rp = 0;
  #pragma unroll
  for (int j = 0; j < 8; ++j) {
    int kk = 4 * (j >> 1) + 2 * h + (j & 1);
    s0p  += sC1s[kk] * Bl1[j];  t0p  += Bl1[j];
    s0rp += sC2s[kk] * Bu2[j];  t0rp += Bu2[j];
  }
  float q0  = (s0p  + __shfl_xor(s0p,  16, 32)) / (t0p  + __shfl_xor(t0p,  16, 32));
  float q0r = (s0rp + __shfl_xor(s0rp, 16, 32)) / (t0rp + __shfl_xor(t0rp, 16, 32));

  // First-occurrence argmin of s/t and argmax of sr/tr (rows 8h+v per lane)
  float lmin = 3.4e38f; int lidx = 0;
  float rmax = -3.4e38f; int ridx = 0;
  #pragma unroll
  for (int v = 0; v < 8; ++v) {
    float rl = S[v] / T[v];
    if (rl < lmin) { lmin = rl; lidx = 8 * h + v; }
    float rr = SR[v] / TR[v];
    if (rr > rmax) { rmax = rr; ridx = 8 * h + v; }
  }
  float plmin = __shfl_xor(lmin, 16, 32); int plidx = __shfl_xor(lidx, 16, 32);
  float prmax = __shfl_xor(rmax, 16, 32); int pridx = __shfl_xor(ridx, 16, 32);
  float loMin = h ? plmin : lmin;  int loLi = h ? plidx : lidx;
  float hiMin = h ? lmin : plmin;  int hiLi = h ? lidx : plidx;
  float gmin;  int gli;
  if (loMin <= hiMin) { gmin = loMin; gli = loLi; } else { gmin = hiMin; gli = hiLi; }
  float loMax = h ? prmax : rmax;  int loRi = h ? pridx : ridx;
  float hiMax = h ? rmax : prmax;  int hiRi = h ? ridx : pridx;
  float gmax;  int gri;
  if (loMax >= hiMax) { gmax = loMax; gri = loRi; } else { gmax = hiMax; gri = hiRi; }
  // L_out = min(cummin(s/t), q0): argmin = 0 if q0 <= min ratio, else first argmin
  int Lloc = (q0  <= gmin) ? 0 : gli;
  int Rloc = (q0r >= gmax) ? 0 : gri;

  // left = sum c1[r]*L_U[r] / sum L_U[r] (reference pairs UNSORTED c with sorted cols)
  float nlp = 0, dlp = 0, nrp = 0, drp = 0;
  #pragma unroll
  for (int j = 0; j < 8; ++j) {
    int kk = 4 * (j >> 1) + 2 * h + (j & 1);
    float lu = (kk <= Lloc) ? Bu1[j] : Bl1[j];
    nlp += sC1[kk] * lu;  dlp += lu;
    float rl = (kk <= Rloc) ? Bl2[j] : Bu2[j];
    nrp += sC2[kk] * rl;  drp += rl;
  }
  float nl = nlp + __shfl_xor(nlp, 16, 32);
  float dl = dlp + __shfl_xor(dlp, 16, 32);
  float nr = nrp + __shfl_xor(nrp, 16, 32);
  float dr = drp + __shfl_xor(drp, 16, 32);

  float res = 0.5f * (nl / dl + nr / dr);
  if (h == 0 && samp < N) out[samp] = res;
}

extern "C" void kernel_launch(void* const* d_in, const int* in_sizes, int n_in,
                              void* d_out, int out_size, void* d_ws, size_t ws_size,
                              hipStream_t stream) {
  const float* input = (const float*)d_in[0];   // (N, 4) f32
  const float* frb   = (const float*)d_in[1];   // (16*4*3) f32
  const float* c1    = (const float*)d_in[2];   // (16,) f32
  const float* c2    = (const float*)d_in[3];   // (16,) f32
  float* out = (float*)d_out;
  (void)d_ws; (void)ws_size; (void)n_in; (void)out_size;

  const int N = in_sizes[0] / ANTE;             // 262144
  const int blocks = (N + 127) / 128;           // 256 thr = 8 waves * 16 samples
  t2fls_km_kernel<<<blocks, 256, 0, stream>>>(input, frb, c1, c2, out, N);
}